// GreedyCTCDecoder_59098749993409
// MI455X (gfx1250) — compile-verified
//
#include <hip/hip_runtime.h>
#include <hip/hip_bf16.h>
#include <stdint.h>

// Greedy CTC decode for emission[T, V=128] f32.
// Pass 1: per-row argmax/max via TDM (tensor_load_to_lds) double-buffered LDS tiles.
// Pass 2: run-length dedup + blank mask + path score.
//
// d_out layout (float, 3*T elems): [0:T) = idx (as float), [T:2T) = keep (0/1),
// [2T:3T) = path_score.  Pass 1 stages max_logit in [2T:3T); pass 2 overwrites it
// with the masked score.

#define CTC_V            128
#define ROWS_PER_TILE    16
#define ROW_BYTES        (CTC_V * 4)          // 512
#define ROW_STRIDE_LDS   (ROW_BYTES + 16)     // 528: 4-DWORD pad per row (bank skew)
#define TILE_LDS_BYTES   (ROWS_PER_TILE * ROW_STRIDE_LDS)  // 8448
#define WAVES_PER_BLOCK  8
#define WAVE_LDS_BYTES   (2 * TILE_LDS_BYTES)              // double buffer / wave
#define BLOCK_LDS_BYTES  (WAVES_PER_BLOCK * WAVE_LDS_BYTES) // 135168

typedef unsigned int u32x4 __attribute__((ext_vector_type(4)));
typedef int          i32x4 __attribute__((ext_vector_type(4)));
typedef int          i32x8 __attribute__((ext_vector_type(8)));

// Issue one TDM load: 16x128 f32 tile (rows are contiguous, stride 128 elems)
// from global `src` into LDS byte address `lds_addr`, with 16B LDS padding after
// every 512B row.  Descriptor per CDNA5 ISA.
__device__ __forceinline__ void tdm_load_tile(const float* src,
                                              uint32_t lds_addr,
                                              uint32_t rows_remaining) {
  uint64_t ga = (uint64_t)(uintptr_t)src;
  // TDM descriptors live in SGPRs: force wave-uniform values.
  int w_lds = __builtin_amdgcn_readfirstlane((int)lds_addr);
  int w_ga0 = __builtin_amdgcn_readfirstlane((int)(uint32_t)ga);
  int w_ga1 = __builtin_amdgcn_readfirstlane(
      (int)((((uint32_t)(ga >> 32)) & 0x01FFFFFFu) | 0x80000000u)); // type=2 (image)
  int dim1  = __builtin_amdgcn_readfirstlane((int)rows_remaining);

  // Group 0: count=1 | lds_addr | global_addr[56:0] | type=2
  u32x4 g0 = { 1u, (uint32_t)w_lds, (uint32_t)w_ga0, (uint32_t)w_ga1 };

  // Group 1:
  //  w0: wg_mask=0, data_size=2 (4B), pad_enable=1, pad_interval=6 (128 DW=512B),
  //      pad_amount=3 (4 DW = 16B)
  //  w1: [63:48] tensor_dim0.lo16 = 128
  //  w2: [95:80] tensor_dim1.lo16
  //  w3: [111:96] tensor_dim1.hi16 | [127:112] tile_dim0 = 128
  //  w4: [143:128] tile_dim1 = 16 rows, tile_dim2 = 0
  //  w5: tensor_dim0_stride = 128 elems
  const int w0 = (int)((2u << 16) | (1u << 20) | (6u << 22) | (3u << 25));
  i32x8 g1 = { w0,
               (int)(128u << 16),
               (int)(((uint32_t)dim1 & 0xFFFFu) << 16),
               (int)((((uint32_t)dim1 >> 16) & 0xFFFFu) | (128u << 16)),
               (int)ROWS_PER_TILE,
               (int)CTC_V,
               0, 0 };

  i32x4 z4 = { 0, 0, 0, 0 };
#if defined(__clang_major__) && __clang_major__ >= 23
  i32x8 z8 = { 0, 0, 0, 0, 0, 0, 0, 0 };
  __builtin_amdgcn_tensor_load_to_lds(g0, g1, z4, z4, z8, 0);
#else
  __builtin_amdgcn_tensor_load_to_lds(g0, g1, z4, z4, 0);
#endif
}

// Pass 1: per-row argmax + max.  Each wave owns tiles of 16 rows; 2 lanes per row
// (lane<16 -> cols 0..63, lane>=16 -> cols 64..127), pairwise combine via shfl_xor 16.
__global__ void ctc_pass1(const float* __restrict__ em,
                          float* __restrict__ out, int T) {
  extern __shared__ __align__(16) unsigned char lds[];
  const int lane = (int)(threadIdx.x & 31u);
  const int wave = (int)(threadIdx.x >> 5);
  const int half = lane >> 4;        // 0: cols 0..63, 1: cols 64..127
  const int r    = lane & 15;        // row within tile
  const int NT   = T / ROWS_PER_TILE;

  const int wgid = (int)blockIdx.x * WAVES_PER_BLOCK + wave;
  const int nw   = (int)gridDim.x * WAVES_PER_BLOCK;

  const uint32_t ldsBase  = (uint32_t)(uintptr_t)&lds[0];
  const uint32_t waveBase = (uint32_t)(wave * WAVE_LDS_BYTES);

  int tile = wgid;
  if (tile < NT) {
    tdm_load_tile(em + (size_t)tile * ROWS_PER_TILE * CTC_V,
                  ldsBase + waveBase,
                  (uint32_t)(T - tile * ROWS_PER_TILE));
  }

  int buf = 0;
  for (int t = tile; t < NT; t += nw) {
    const int tn = t + nw;
    if (tn < NT) {
      // Buffer buf^1 was read in the previous iteration; make sure those ds_loads
      // retired before the TDM engine overwrites it.
      asm volatile("s_wait_dscnt 0" ::: "memory");
      tdm_load_tile(em + (size_t)tn * ROWS_PER_TILE * CTC_V,
                    ldsBase + waveBase + (uint32_t)((buf ^ 1) * TILE_LDS_BYTES),
                    (uint32_t)(T - tn * ROWS_PER_TILE));
      __builtin_amdgcn_s_wait_tensorcnt(1);   // oldest (tile t) complete
    } else {
      __builtin_amdgcn_s_wait_tensorcnt(0);
    }
    asm volatile("" ::: "memory");

    const unsigned char* rowBase =
        lds + waveBase + buf * TILE_LDS_BYTES + r * ROW_STRIDE_LDS + half * (ROW_BYTES / 2);

    float bestV = -3.402823466e+38f;
    int   bestI = 0;
#pragma unroll
    for (int k = 0; k < 16; ++k) {
      const float4 v = *(const float4*)(rowBase + k * 16);
      const int i0 = (half << 6) + (k << 2);
      if (v.x > bestV) { bestV = v.x; bestI = i0;     }
      if (v.y > bestV) { bestV = v.y; bestI = i0 + 1; }
      if (v.z > bestV) { bestV = v.z; bestI = i0 + 2; }
      if (v.w > bestV) { bestV = v.w; bestI = i0 + 3; }
    }
    // Combine the two half-rows; ties -> lower index (first occurrence, like argmax).
    const float oV = __shfl_xor(bestV, 16, 32);
    const int   oI = __shfl_xor(bestI, 16, 32);
    if (oV > bestV || (oV == bestV && oI < bestI)) { bestV = oV; bestI = oI; }

    if (half == 0) {
      const int row = t * ROWS_PER_TILE + r;
      out[row]                   = (float)bestI;   // idx
      out[(size_t)2 * T + row]   = bestV;          // max_logit (staged)
    }
    buf ^= 1;
  }
}

// Pass 2: keep + path_score.
__global__ void ctc_pass2(float* __restrict__ out, int T) {
  const int t = (int)(blockIdx.x * blockDim.x + threadIdx.x);
  if (t >= T) return;
  const float idx  = out[t];
  const float prev = (t == 0) ? -1.0f : out[t - 1];
  const float ml   = out[(size_t)2 * T + t];
  const bool keep  = (idx != prev) && (idx != 0.0f);   // BLANK == 0
  out[(size_t)T + t]     = keep ? 1.0f : 0.0f;
  out[(size_t)2 * T + t] = keep ? ml : 0.0f;
}

extern "C" void kernel_launch(void* const* d_in, const int* in_sizes, int n_in,
                              void* d_out, int out_size, void* d_ws, size_t ws_size,
                              hipStream_t stream) {
  (void)n_in; (void)out_size; (void)d_ws; (void)ws_size;
  const float* em = (const float*)d_in[0];
  float* out = (float*)d_out;
  const int T = in_sizes[0] / CTC_V;   // 1048576 (multiple of 16)

  // Pass 1: 1024 blocks x 8 waves = 8192 waves; 65536 tiles -> 8 tiles/wave,
  // double-buffered TDM keeps 2 x 8.4KB in flight per wave.
  ctc_pass1<<<1024, WAVES_PER_BLOCK * 32, BLOCK_LDS_BYTES, stream>>>(em, out, T);

  // Pass 2: elementwise.
  ctc_pass2<<<(T + 255) / 256, 256, 0, stream>>>(out, T);
}